// CausalSelfAttention_44083544326734
// MI455X (gfx1250) — compile-verified
//
#include <hip/hip_runtime.h>
#include <hip/hip_bf16.h>

// CausalSelfAttention with relative-position rank-2 bias, fp32 WMMA (gfx1250).
// rel@Wrk / rel@Wrv tensors are algebraically collapsed to rank-2 terms.

typedef __attribute__((ext_vector_type(2))) float v2f;
typedef __attribute__((ext_vector_type(8))) float v8f;

#define B_  2
#define T_  768
#define H_  512
#define NH_ 8
#define HD_ 64
#define SCALE 0.125f   // 1/sqrt(64)

// ---- workspace layout (float offsets) ----
#define Q_OFF   ((size_t)0)
#define K_OFF   (Q_OFF  + (size_t)B_*NH_*T_*HD_)   // 786432
#define V_OFF   (K_OFF  + (size_t)B_*NH_*T_*HD_)
#define W_OFF   (V_OFF  + (size_t)B_*NH_*T_*HD_)   // [B,T,H] attn output pre-proj
#define A0_OFF  (W_OFF  + (size_t)B_*T_*H_)        // q.Wrk0 per (bh,t)
#define A1_OFF  (A0_OFF + (size_t)B_*NH_*T_)
#define A2_OFF  (A1_OFF + (size_t)B_*NH_*T_)       // q.brk
#define C0_OFF  (A2_OFF + (size_t)B_*NH_*T_)       // sum p*dist
#define C1_OFF  (C0_OFF + (size_t)B_*NH_*T_)       // sum p*tint
#define S_OFF   (C1_OFF + (size_t)B_*NH_*T_)       // scores/probs [bh][t][s]

__device__ __forceinline__ v8f wmma_f32(v2f a, v2f b, v8f c) {
  return __builtin_amdgcn_wmma_f32_16x16x4_f32(false, a, false, b, (short)0, c,
                                               false, false);
}

// ---- Kernel 1: qkv = x @ Wqkv + bqkv, scattered to q/k/v [b,h,t,d] ----
// grid (48,24), block 256. 8 waves/block in 2(M)x4(N); wave tile 16x16.
__global__ void qkv_gemm(const float* __restrict__ x, const float* __restrict__ Wqkv,
                         const float* __restrict__ bqkv, float* __restrict__ ws) {
  const int lane = threadIdx.x & 31, wid = threadIdx.x >> 5;
  const int hi = lane >> 4, ln = lane & 15;
  const int m_base = blockIdx.x * 32 + (wid & 1) * 16;
  const int n_base = blockIdx.y * 64 + (wid >> 1) * 16;
  const int m_row = m_base + ln;
  const int n_col = n_base + ln;

  v8f acc;
  const float bias = bqkv[n_col];
#pragma unroll
  for (int r = 0; r < 8; ++r) acc[r] = bias;

  const float* arow = x + (size_t)m_row * H_;
  for (int kb = 0; kb < H_; kb += 4) {
    v2f a = *(const v2f*)(arow + kb + 2 * hi);
    int kk = kb + 2 * hi;
    v2f bf;
    bf.x = Wqkv[(size_t)kk * (3 * H_) + n_col];
    bf.y = Wqkv[(size_t)(kk + 1) * (3 * H_) + n_col];
    acc = wmma_f32(a, bf, acc);
  }

  const int which = n_col / H_;           // wave-uniform (64-col tiles)
  const int w_in = n_col % H_;
  const int h = w_in / HD_, d = w_in % HD_;
  float* dst = ws + (which == 0 ? Q_OFF : (which == 1 ? K_OFF : V_OFF));
#pragma unroll
  for (int r = 0; r < 8; ++r) {
    int m = m_base + r + 8 * hi;
    int b = m / T_, t = m % T_;
    dst[(((size_t)b * NH_ + h) * T_ + t) * HD_ + d] = acc[r];
  }
}

// ---- Kernel 2: a0 = q.Wrk0, a1 = q.Wrk1, a2 = q.brk per (bh,t) ----
__global__ void rel_q_proj(const float* __restrict__ Wrk, const float* __restrict__ brk,
                           float* __restrict__ ws) {
  int idx = blockIdx.x * blockDim.x + threadIdx.x;
  if (idx >= B_ * NH_ * T_) return;
  const float* qrow = ws + Q_OFF + (size_t)idx * HD_;
  float a0 = 0.f, a1 = 0.f, a2 = 0.f;
#pragma unroll 8
  for (int d = 0; d < HD_; ++d) {
    float qd = qrow[d];
    a0 += qd * Wrk[d];
    a1 += qd * Wrk[HD_ + d];
    a2 += qd * brk[d];
  }
  ws[A0_OFF + idx] = a0;
  ws[A1_OFF + idx] = a1;
  ws[A2_OFF + idx] = a2;
}

// ---- Kernel 3: S = (q.kT + a0*dist + a1*tint + a2)*scale, masked ----
// grid (288, 16=bh), block 256; each wave does one 16x16 (t,s) tile.
__global__ void scores_kernel(const float* __restrict__ dist, const float* __restrict__ tint,
                              const int* __restrict__ tlen, float* __restrict__ ws) {
  const int lane = threadIdx.x & 31, wid = threadIdx.x >> 5;
  const int hi = lane >> 4, ln = lane & 15;
  const int bh = blockIdx.y;
  const int tile = blockIdx.x * 8 + wid;          // 0..2303
  const int tt = tile / (T_ / 16), ss = tile % (T_ / 16);
  const int t0 = tt * 16, s0 = ss * 16;
  const int b = bh / NH_;
  float* S = ws + S_OFF + (size_t)bh * T_ * T_;

  if (s0 > t0 + 15) {                             // wave-uniform: above diagonal
#pragma unroll
    for (int r = 0; r < 8; ++r) {
      int t = t0 + r + 8 * hi;
      S[(size_t)t * T_ + s0 + ln] = -__builtin_inff();
    }
    return;
  }

  const float* qrow = ws + Q_OFF + ((size_t)bh * T_ + (t0 + ln)) * HD_;
  const float* krow = ws + K_OFF + ((size_t)bh * T_ + (s0 + ln)) * HD_;
  v8f acc = {0.f, 0.f, 0.f, 0.f, 0.f, 0.f, 0.f, 0.f};
#pragma unroll
  for (int kb = 0; kb < HD_; kb += 4) {
    v2f a = *(const v2f*)(qrow + kb + 2 * hi);
    v2f bf = *(const v2f*)(krow + kb + 2 * hi);
    acc = wmma_f32(a, bf, acc);
  }

  const int len = tlen[b];
#pragma unroll
  for (int r = 0; r < 8; ++r) {
    int t = t0 + r + 8 * hi;
    int s = s0 + ln;
    float a0 = ws[A0_OFF + (size_t)bh * T_ + t];
    float a1 = ws[A1_OFF + (size_t)bh * T_ + t];
    float a2 = ws[A2_OFF + (size_t)bh * T_ + t];
    size_t di = ((size_t)b * T_ + t) * T_ + s;
    float val = (acc[r] + a0 * dist[di] + a1 * tint[di] + a2) * SCALE;
    if (s > t || s >= len) val = -__builtin_inff();
    S[(size_t)t * T_ + s] = val;
  }
}

// ---- Kernel 4: rowwise softmax in-place; also c0 = sum p*dist, c1 = sum p*tint ----
// grid 1536, block 256; one wave per (bh,t) row of 768.
__global__ void softmax_kernel(const float* __restrict__ dist, const float* __restrict__ tint,
                               float* __restrict__ ws) {
  const int lane = threadIdx.x & 31, wid = threadIdx.x >> 5;
  const int row = blockIdx.x * 8 + wid;           // (bh,t)
  const int bh = row / T_, t = row % T_;
  const int b = bh / NH_;
  float* S = ws + S_OFF + ((size_t)bh * T_ + t) * T_;
  const float* drow = dist + ((size_t)b * T_ + t) * T_;
  const float* trow = tint + ((size_t)b * T_ + t) * T_;

  float m = -__builtin_inff();
  for (int s = lane; s < T_; s += 32) m = fmaxf(m, S[s]);
#pragma unroll
  for (int off = 16; off > 0; off >>= 1) m = fmaxf(m, __shfl_xor(m, off, 32));

  const bool dead = !(m > -3.0e38f);              // fully-masked row guard
  float pv[T_ / 32];
  float sum = 0.f;
  int i = 0;
  for (int s = lane; s < T_; s += 32, ++i) {
    float p = dead ? 0.f : __expf(S[s] - m);
    pv[i] = p;
    sum += p;
  }
#pragma unroll
  for (int off = 16; off > 0; off >>= 1) sum += __shfl_xor(sum, off, 32);
  const float inv = sum > 0.f ? 1.f / sum : 0.f;

  float c0 = 0.f, c1 = 0.f;
  i = 0;
  for (int s = lane; s < T_; s += 32, ++i) {
    float p = pv[i] * inv;
    S[s] = p;
    c0 += p * drow[s];
    c1 += p * trow[s];
  }
#pragma unroll
  for (int off = 16; off > 0; off >>= 1) {
    c0 += __shfl_xor(c0, off, 32);
    c1 += __shfl_xor(c1, off, 32);
  }
  if (lane == 0) {
    ws[C0_OFF + row] = c0;
    ws[C1_OFF + row] = c1;
  }
}

// ---- Kernel 5: w = P@V + c0*Wrv0 + c1*Wrv1 + brv, into [B,T,H] ----
// grid 384, block 256; wave = (bh, t-tile, d-tile); k-loop truncated by causality.
__global__ void attnv_kernel(const float* __restrict__ Wrv, const float* __restrict__ brv,
                             float* __restrict__ ws) {
  const int lane = threadIdx.x & 31, wid = threadIdx.x >> 5;
  const int hi = lane >> 4, ln = lane & 15;
  const int task = blockIdx.x * 2 + (wid >> 2);   // (bh, tt) pairs
  const int nt = wid & 3;
  const int bh = task / (T_ / 16), tt = task % (T_ / 16);
  const int t0 = tt * 16, d0 = nt * 16;

  const float* prow = ws + S_OFF + (size_t)bh * T_ * T_ + (size_t)(t0 + ln) * T_;
  const float* Vm = ws + V_OFF + (size_t)bh * T_ * HD_;
  v8f acc = {0.f, 0.f, 0.f, 0.f, 0.f, 0.f, 0.f, 0.f};

  const int smax = t0 + 16;                       // p==0 beyond diagonal block
  for (int kb = 0; kb < smax; kb += 4) {
    v2f a = *(const v2f*)(prow + kb + 2 * hi);
    int kk = kb + 2 * hi;
    v2f bf;
    bf.x = Vm[(size_t)kk * HD_ + d0 + ln];
    bf.y = Vm[(size_t)(kk + 1) * HD_ + d0 + ln];
    acc = wmma_f32(a, bf, acc);
  }

  const int b = bh / NH_, h = bh % NH_;
  const int d = d0 + ln;
  const float w0 = Wrv[d], w1 = Wrv[HD_ + d], bv = brv[d];
  float* wout = ws + W_OFF;
#pragma unroll
  for (int r = 0; r < 8; ++r) {
    int t = t0 + r + 8 * hi;
    float c0 = ws[C0_OFF + (size_t)bh * T_ + t];
    float c1 = ws[C1_OFF + (size_t)bh * T_ + t];
    wout[((size_t)b * T_ + t) * H_ + h * HD_ + d] = acc[r] + c0 * w0 + c1 * w1 + bv;
  }
}

// ---- Kernel 6: out = w @ Wo + bo ----
// grid (48,8), block 256. Same structure as qkv_gemm, N=512.
__global__ void out_proj(const float* __restrict__ Wo, const float* __restrict__ bo,
                         const float* __restrict__ ws, float* __restrict__ out) {
  const int lane = threadIdx.x & 31, wid = threadIdx.x >> 5;
  const int hi = lane >> 4, ln = lane & 15;
  const int m_base = blockIdx.x * 32 + (wid & 1) * 16;
  const int n_base = blockIdx.y * 64 + (wid >> 1) * 16;
  const int m_row = m_base + ln;
  const int n_col = n_base + ln;

  v8f acc;
  const float bias = bo[n_col];
#pragma unroll
  for (int r = 0; r < 8; ++r) acc[r] = bias;

  const float* arow = ws + W_OFF + (size_t)m_row * H_;
  for (int kb = 0; kb < H_; kb += 4) {
    v2f a = *(const v2f*)(arow + kb + 2 * hi);
    int kk = kb + 2 * hi;
    v2f bf;
    bf.x = Wo[(size_t)kk * H_ + n_col];
    bf.y = Wo[(size_t)(kk + 1) * H_ + n_col];
    acc = wmma_f32(a, bf, acc);
  }
#pragma unroll
  for (int r = 0; r < 8; ++r) {
    int m = m_base + r + 8 * hi;
    out[(size_t)m * H_ + n_col] = acc[r];
  }
}

extern "C" void kernel_launch(void* const* d_in, const int* in_sizes, int n_in,
                              void* d_out, int out_size, void* d_ws, size_t ws_size,
                              hipStream_t stream) {
  const float* x    = (const float*)d_in[0];
  const float* dist = (const float*)d_in[1];
  const float* tint = (const float*)d_in[2];
  const int*   tlen = (const int*)d_in[3];
  const float* Wqkv = (const float*)d_in[4];
  const float* bqkv = (const float*)d_in[5];
  const float* Wrk  = (const float*)d_in[6];
  const float* brk  = (const float*)d_in[7];
  const float* Wrv  = (const float*)d_in[8];
  const float* brv  = (const float*)d_in[9];
  const float* Wo   = (const float*)d_in[10];
  const float* bo   = (const float*)d_in[11];
  float* out = (float*)d_out;
  float* ws  = (float*)d_ws;

  qkv_gemm<<<dim3(48, 24), 256, 0, stream>>>(x, Wqkv, bqkv, ws);
  rel_q_proj<<<48, 256, 0, stream>>>(Wrk, brk, ws);
  scores_kernel<<<dim3(288, 16), 256, 0, stream>>>(dist, tint, tlen, ws);
  softmax_kernel<<<1536, 256, 0, stream>>>(dist, tint, ws);
  attnv_kernel<<<384, 256, 0, stream>>>(Wrv, brv, ws);
  out_proj<<<dim3(48, 8), 256, 0, stream>>>(Wo, bo, ws, out);
}